// RoINet_76922864271896
// MI455X (gfx1250) — compile-verified
//
#include <hip/hip_runtime.h>
#include <stdint.h>

#define NUM_FG   80
#define A_ANCH   9
#define PRE_NMS  1000
#define TOP_N    100
#define NMS_THR  0.5f
#define IMG_SZ   640.0f
#define LOG_MAX_ 4.135166556742356f
#define RPI      3000   // rois per image = 3 levels * 1000

typedef __attribute__((ext_vector_type(2))) float v2f;
typedef __attribute__((ext_vector_type(8))) float v8f;

#if defined(__has_builtin)
#  if __has_builtin(__builtin_amdgcn_wmma_f32_16x16x4_f32)
#    define HAVE_WMMA_F32X4 1
#  endif
#  if __has_builtin(__builtin_amdgcn_global_load_async_to_lds_b32)
#    define HAVE_ASYNC_LDS 1
#  endif
#  if __has_builtin(__builtin_amdgcn_s_wait_asynccnt)
#    define HAVE_WAIT_ASYNC 1
#  endif
#endif
#ifndef HAVE_WMMA_F32X4
#  define HAVE_WMMA_F32X4 0
#endif
#ifndef HAVE_ASYNC_LDS
#  define HAVE_ASYNC_LDS 0
#endif
#ifndef HAVE_WAIT_ASYNC
#  define HAVE_WAIT_ASYNC 0
#endif

__device__ __forceinline__ void wait_async_lds() {
#if HAVE_WAIT_ASYNC
  __builtin_amdgcn_s_wait_asynccnt(0);
#else
  asm volatile("s_wait_asynccnt 0" ::: "memory");
#endif
}

// Order-preserving float->uint key: larger float => larger key.
__device__ __forceinline__ uint32_t fkey(float x) {
  uint32_t u = __float_as_uint(x);
  return (u & 0x80000000u) ? ~u : (u | 0x80000000u);
}

// ---------------------------------------------------------------------------
// Pass 1 histogram: 4096 bins over the top 12 bits of the key, per image.
// cls layout per image is contiguous: [A*NUM_FG, hh, ww] = N floats.
// ---------------------------------------------------------------------------
__global__ __launch_bounds__(256) void hist_pass1(const float* __restrict__ cls,
                                                  uint32_t* __restrict__ hist,
                                                  int N) {
  __shared__ uint32_t lh[4096];
  const int b = blockIdx.y;
  const float* cp = cls + (size_t)b * N;
  uint32_t* gh = hist + (size_t)b * 4096;
  for (int i = threadIdx.x; i < 4096; i += blockDim.x) lh[i] = 0;
  __syncthreads();
  const int stride = gridDim.x * blockDim.x;
  const int n4 = N >> 2;
  for (int i = blockIdx.x * blockDim.x + threadIdx.x; i < n4; i += stride) {
    float4 v = ((const float4*)cp)[i];
    atomicAdd(&lh[fkey(v.x) >> 20], 1u);
    atomicAdd(&lh[fkey(v.y) >> 20], 1u);
    atomicAdd(&lh[fkey(v.z) >> 20], 1u);
    atomicAdd(&lh[fkey(v.w) >> 20], 1u);
  }
  __syncthreads();
  for (int i = threadIdx.x; i < 4096; i += blockDim.x)
    if (lh[i]) atomicAdd(&gh[i], lh[i]);
}

// ---------------------------------------------------------------------------
// Pass 2 histogram: only elements whose top-12 key bits == threshold bin;
// histogram the next 12 key bits (bits 19..8).
// ---------------------------------------------------------------------------
__global__ __launch_bounds__(256) void hist_pass2(const float* __restrict__ cls,
                                                  const uint32_t* __restrict__ thrInfo,
                                                  uint32_t* __restrict__ hist2,
                                                  int N) {
  __shared__ uint32_t lh[4096];
  const int b = blockIdx.y;
  const uint32_t bin1 = thrInfo[(size_t)b * 8 + 0];
  const float* cp = cls + (size_t)b * N;
  uint32_t* gh = hist2 + (size_t)b * 4096;
  for (int i = threadIdx.x; i < 4096; i += blockDim.x) lh[i] = 0;
  __syncthreads();
  const int stride = gridDim.x * blockDim.x;
  const int n4 = N >> 2;
  for (int i = blockIdx.x * blockDim.x + threadIdx.x; i < n4; i += stride) {
    float4 v = ((const float4*)cp)[i];
    uint32_t k;
    k = fkey(v.x); if ((k >> 20) == bin1) atomicAdd(&lh[(k >> 8) & 0xFFFu], 1u);
    k = fkey(v.y); if ((k >> 20) == bin1) atomicAdd(&lh[(k >> 8) & 0xFFFu], 1u);
    k = fkey(v.z); if ((k >> 20) == bin1) atomicAdd(&lh[(k >> 8) & 0xFFFu], 1u);
    k = fkey(v.w); if ((k >> 20) == bin1) atomicAdd(&lh[(k >> 8) & 0xFFFu], 1u);
  }
  __syncthreads();
  for (int i = threadIdx.x; i < 4096; i += blockDim.x)
    if (lh[i]) atomicAdd(&gh[i], lh[i]);
}

// ---------------------------------------------------------------------------
// Radix-select: find the largest bin T with suffix_count(T) >= target.
// Suffix sums over 4096 bins are computed EXACTLY in f32 (counts < 2^24) via
// WMMA: D = T_upper_triangular(16x16, ones) x H(16x16 tile), chained over K=4
// slices of V_WMMA_F32_16X16X4_F32. One wave per (image).
//   pass==0: target=1000, writes info[0]=thrBin1, info[1]=strictAbove1
//   pass==1: target=1000-info[1], writes info[2..6]
//            info[4]=thr24 (top-24-bit key threshold), info[5]=numTakeEq,
//            info[6]=count of keys strictly above thr24.
// ---------------------------------------------------------------------------
__global__ __launch_bounds__(32) void radix_select(const uint32_t* __restrict__ hist,
                                                   uint32_t* __restrict__ thrInfo,
                                                   int pass) {
  const int b = blockIdx.x;
  const uint32_t* h = hist + (size_t)b * 4096;
  uint32_t* info = thrInfo + (size_t)b * 8;

  __shared__ float suf[4096];   // within-group (16) inclusive suffix per bin
  __shared__ float gtot[256];   // group totals
  __shared__ float ws2s[256];   // within-supergroup inclusive suffix of gtot
  __shared__ float csuf[16];    // strict suffix over supergroup totals
  __shared__ float sgAdd[256];  // strictly-above-group count per group
  __shared__ int   bestArr[32];

  const int lane = threadIdx.x;       // single wave32
  const int n = lane & 15;            // A row / B,D column
  const bool hi = lane >= 16;

  const float target = (pass == 0) ? (float)PRE_NMS : (float)(PRE_NMS - (int)info[1]);

#if HAVE_WMMA_F32X4
  // ---- level 1: 16 tiles, each tile = 16 groups (columns) x 16 elems (rows)
  for (int tile = 0; tile < 16; ++tile) {
    v8f acc = {0.f, 0.f, 0.f, 0.f, 0.f, 0.f, 0.f, 0.f};
#pragma unroll
    for (int c = 0; c < 4; ++c) {
      const int e0 = c * 4 + (hi ? 2 : 0);
      v2f a, bm;
      a.x = (e0 >= n) ? 1.0f : 0.0f;       // T[n][e0]
      a.y = (e0 + 1 >= n) ? 1.0f : 0.0f;   // T[n][e0+1]
      const int g = tile * 16 + n;         // column = group
      bm.x = (float)h[g * 16 + e0];
      bm.y = (float)h[g * 16 + e0 + 1];
      acc = __builtin_amdgcn_wmma_f32_16x16x4_f32(false, a, false, bm,
                                                  (short)0, acc, false, false);
    }
    const int g = tile * 16 + n;
#pragma unroll
    for (int r = 0; r < 8; ++r)
      suf[g * 16 + r + (hi ? 8 : 0)] = acc[r];   // D[M=row e][N=group]
  }
  __syncthreads();
  for (int g = lane; g < 256; g += 32) gtot[g] = suf[g * 16];  // e=0 => full sum
  __syncthreads();
  // ---- level 2: one tile over the 256 group totals (16 supergroups x 16)
  {
    v8f acc = {0.f, 0.f, 0.f, 0.f, 0.f, 0.f, 0.f, 0.f};
#pragma unroll
    for (int c = 0; c < 4; ++c) {
      const int e0 = c * 4 + (hi ? 2 : 0);
      v2f a, bm;
      a.x = (e0 >= n) ? 1.0f : 0.0f;
      a.y = (e0 + 1 >= n) ? 1.0f : 0.0f;
      bm.x = gtot[n * 16 + e0];
      bm.y = gtot[n * 16 + e0 + 1];
      acc = __builtin_amdgcn_wmma_f32_16x16x4_f32(false, a, false, bm,
                                                  (short)0, acc, false, false);
    }
#pragma unroll
    for (int r = 0; r < 8; ++r)
      ws2s[n * 16 + r + (hi ? 8 : 0)] = acc[r];
  }
  __syncthreads();
#else
  // Scalar fallback with identical semantics
  for (int g = lane; g < 256; g += 32) {
    float s = 0.0f;
    for (int e = 15; e >= 0; --e) { s += (float)h[g * 16 + e]; suf[g * 16 + e] = s; }
    gtot[g] = s;
  }
  __syncthreads();
  if (lane < 16) {
    float s = 0.0f;
    for (int k = 15; k >= 0; --k) { s += gtot[lane * 16 + k]; ws2s[lane * 16 + k] = s; }
  }
  __syncthreads();
#endif

  if (lane < 16) {
    float s = 0.0f;
    for (int n2 = lane + 1; n2 < 16; ++n2) s += ws2s[n2 * 16];  // supergroup totals
    csuf[lane] = s;
  }
  __syncthreads();
  for (int G = lane; G < 256; G += 32)
    sgAdd[G] = ws2s[G] + csuf[G >> 4] - gtot[G];   // strictly above group G
  __syncthreads();

  int best = -1;
  for (int bin = lane; bin < 4096; bin += 32) {
    float fs = suf[bin] + sgAdd[bin >> 4];         // inclusive suffix count
    if (fs >= target && bin > best) best = bin;
  }
  bestArr[lane] = best;
  __syncthreads();
  if (lane == 0) {
    int thr = -1;
    for (int i = 0; i < 32; ++i) thr = max(thr, bestArr[i]);
    if (thr < 0) thr = 0;
    const float fs = suf[thr] + sgAdd[thr >> 4];
    const uint32_t strictAbove = (uint32_t)rintf(fs - (float)h[thr]);
    if (pass == 0) {
      info[0] = (uint32_t)thr;
      info[1] = strictAbove;
    } else {
      info[2] = (uint32_t)thr;
      info[3] = strictAbove;
      const uint32_t sA1 = info[1];
      const uint32_t tgt2 = (uint32_t)PRE_NMS - sA1;
      info[4] = (info[0] << 12) | (uint32_t)thr;   // thr24
      info[5] = tgt2 - strictAbove;                // numTakeEq
      info[6] = sA1 + strictAbove;                 // count strictly > thr24
    }
  }
}

// ---------------------------------------------------------------------------
// Compact winners + sigmoid + anchor box decode, writing 6-float rois.
// ---------------------------------------------------------------------------
__global__ __launch_bounds__(256) void compact_decode(const float* __restrict__ cls,
                                                      const float* __restrict__ loc,
                                                      const uint32_t* __restrict__ thrInfo,
                                                      uint32_t* __restrict__ counters,
                                                      float* __restrict__ rois,
                                                      int hh, int ww, int strideA,
                                                      int lvlOff) {
  const int b = blockIdx.y;
  const int plane = hh * ww;
  const int N = NUM_FG * A_ANCH * plane;
  const float* cp = cls + (size_t)b * N;
  const uint32_t* info = thrInfo + (size_t)b * 8;
  const uint32_t thr24 = info[4];
  const uint32_t numTakeEq = info[5];
  const uint32_t cntGt = info[6];
  uint32_t* cnt = counters + (size_t)b * 2;

  const int gstride = gridDim.x * blockDim.x;
  for (int e = blockIdx.x * blockDim.x + threadIdx.x; e < N; e += gstride) {
    const float logit = cp[e];
    const uint32_t k24 = fkey(logit) >> 8;
    if (k24 < thr24) continue;
    uint32_t slot;
    if (k24 > thr24) {
      slot = atomicAdd(&cnt[0], 1u);
    } else {
      const uint32_t t = atomicAdd(&cnt[1], 1u);
      if (t >= numTakeEq) continue;
      slot = cntGt + t;
    }
    if (slot >= (uint32_t)PRE_NMS) continue;  // safety clamp

    const int x = e % ww;
    const int y = (e / ww) % hh;
    const int ch = e / plane;
    const int a = ch / NUM_FG;
    const int f = ch - a * NUM_FG;

    // ---- anchor (classic py-faster-rcnn, np.round == rintf half-to-even)
    const float ratios[3] = {0.5f, 1.0f, 2.0f};
    const float scales[3] = {4.0f, 8.0f, 16.0f};
    const float ratio = ratios[a / 3];
    const float scale = scales[a - (a / 3) * 3];
    const float base = (float)strideA;
    const float ctr = (base - 1.0f) * 0.5f;
    const float ws0 = rintf(sqrtf(base * base / ratio));
    const float hs0 = rintf(ws0 * ratio);
    const float wA = ws0 * scale;
    const float hA = hs0 * scale;
    const float cxA = ctr + (float)(x * strideA);
    const float cyA = ctr + (float)(y * strideA);

    // ---- deltas: loc[b, a*4+c, y, x]
    const float* lp = loc + ((((size_t)b * (A_ANCH * 4) + (size_t)a * 4) * hh + y) * ww + x);
    const float dx = lp[0];
    const float dy = lp[(size_t)plane];
    float dw = lp[(size_t)2 * plane];
    float dh = lp[(size_t)3 * plane];
    dw = fminf(fmaxf(dw, -LOG_MAX_), LOG_MAX_);
    dh = fminf(fmaxf(dh, -LOG_MAX_), LOG_MAX_);

    const float pcx = dx * wA + cxA;
    const float pcy = dy * hA + cyA;
    const float pw = expf(dw) * wA;
    const float ph = expf(dh) * hA;
    float x1 = pcx - 0.5f * (pw - 1.0f);
    float y1 = pcy - 0.5f * (ph - 1.0f);
    float x2 = pcx + 0.5f * (pw - 1.0f);
    float y2 = pcy + 0.5f * (ph - 1.0f);
    x1 = fminf(fmaxf(x1, 0.0f), IMG_SZ - 1.0f);
    y1 = fminf(fmaxf(y1, 0.0f), IMG_SZ - 1.0f);
    x2 = fminf(fmaxf(x2, 0.0f), IMG_SZ - 1.0f);
    y2 = fminf(fmaxf(y2, 0.0f), IMG_SZ - 1.0f);

    const float score = 1.0f / (1.0f + expf(-logit));
    float* o = rois + ((size_t)b * RPI + lvlOff + slot) * 6;
    o[0] = x1; o[1] = y1; o[2] = x2; o[3] = y2;
    o[4] = score; o[5] = (float)(f + 1);
  }
}

// ---------------------------------------------------------------------------
// Per-image bitonic sort of 3000 rois by score descending (pad to 4096).
// Scores are staged into LDS with async global->LDS loads on CDNA5.
// ---------------------------------------------------------------------------
__global__ __launch_bounds__(1024) void sort_rois(const float* __restrict__ rois,
                                                  float* __restrict__ srois) {
  __shared__ float key[4096];
  __shared__ int   sidx[4096];
  const int b = blockIdx.x;
  const int tid = threadIdx.x;
  const float* rb = rois + (size_t)b * RPI * 6;

  for (int i = tid; i < 4096; i += 1024) {
    sidx[i] = i;
    if (i < RPI) {
#if HAVE_ASYNC_LDS
      __builtin_amdgcn_global_load_async_to_lds_b32(
          (__attribute__((address_space(1))) int*)(rb + (size_t)i * 6 + 4),
          (__attribute__((address_space(3))) int*)&key[i], 0, 0);
#else
      key[i] = rb[(size_t)i * 6 + 4];
#endif
    } else {
      key[i] = -__builtin_inff();
    }
  }
#if HAVE_ASYNC_LDS
  wait_async_lds();
#endif
  __syncthreads();

  for (int k = 2; k <= 4096; k <<= 1) {
    for (int j = k >> 1; j > 0; j >>= 1) {
      for (int i = tid; i < 4096; i += 1024) {
        const int ixj = i ^ j;
        if (ixj > i) {
          const bool up = (i & k) == 0;
          const float a = key[i], c = key[ixj];
          const bool sw = up ? (a < c) : (a > c);   // overall descending
          if (sw) {
            key[i] = c; key[ixj] = a;
            const int t = sidx[i]; sidx[i] = sidx[ixj]; sidx[ixj] = t;
          }
        }
      }
      __syncthreads();
    }
  }

  for (int t = tid; t < RPI; t += 1024) {
    const int s = sidx[t];
    const float* src = rb + (size_t)s * 6;
    float* dst = srois + ((size_t)b * RPI + t) * 6;
#pragma unroll
    for (int c = 0; c < 6; ++c) dst[c] = src[c];
  }
}

// ---------------------------------------------------------------------------
// Greedy NMS on class-offset boxes (box + cls*(IMG+1)) + emit top-100 rows.
// One 32-wave workgroup per image; boxes live in LDS (~60KB static).
// ---------------------------------------------------------------------------
__global__ __launch_bounds__(1024) void nms_out(const float* __restrict__ srois,
                                                float* __restrict__ out) {
  __shared__ float bx1[RPI], by1[RPI], bx2[RPI], by2[RPI];
  __shared__ uint32_t flag[RPI];
  __shared__ int keep[TOP_N];
  __shared__ int nkeep;
  const int b = blockIdx.x;
  const int tid = threadIdx.x;
  const float* rb = srois + (size_t)b * RPI * 6;

  for (int i = tid; i < RPI; i += 1024) {
    const float off = rb[(size_t)i * 6 + 5] * (IMG_SZ + 1.0f);
    bx1[i] = rb[(size_t)i * 6 + 0] + off;
    by1[i] = rb[(size_t)i * 6 + 1] + off;
    bx2[i] = rb[(size_t)i * 6 + 2] + off;
    by2[i] = rb[(size_t)i * 6 + 3] + off;
    flag[i] = 0;
  }
  __syncthreads();

  for (int i = 0; i < RPI - 1; ++i) {
    if (flag[i] == 0) {   // uniform across block
      const float ax1 = bx1[i], ay1 = by1[i], ax2 = bx2[i], ay2 = by2[i];
      const float aarea = (ax2 - ax1 + 1.0f) * (ay2 - ay1 + 1.0f);
      for (int j = i + 1 + tid; j < RPI; j += 1024) {
        if (flag[j]) continue;
        const float xx1 = fmaxf(ax1, bx1[j]);
        const float yy1 = fmaxf(ay1, by1[j]);
        const float xx2 = fminf(ax2, bx2[j]);
        const float yy2 = fminf(ay2, by2[j]);
        const float w = fmaxf(xx2 - xx1 + 1.0f, 0.0f);
        const float h = fmaxf(yy2 - yy1 + 1.0f, 0.0f);
        const float inter = w * h;
        const float barea = (bx2[j] - bx1[j] + 1.0f) * (by2[j] - by1[j] + 1.0f);
        const float iou = inter / (aarea + barea - inter);
        if (iou > NMS_THR) flag[j] = 1;
      }
    }
    __syncthreads();
  }

  if (tid == 0) {
    int nk = 0;
    for (int t = 0; t < RPI && nk < TOP_N; ++t)
      if (!flag[t]) keep[nk++] = t;
    nkeep = nk;
  }
  __syncthreads();

  for (int t = tid; t < TOP_N; t += 1024) {
    float* o = out + ((size_t)b * TOP_N + t) * 7;
    if (t < nkeep) {
      const float* r = rb + (size_t)keep[t] * 6;
      o[0] = (float)b;
#pragma unroll
      for (int c = 0; c < 6; ++c) o[1 + c] = r[c];
    } else {
#pragma unroll
      for (int c = 0; c < 7; ++c) o[c] = 0.0f;
    }
  }
}

// ---------------------------------------------------------------------------
extern "C" void kernel_launch(void* const* d_in, const int* in_sizes, int n_in,
                              void* d_out, int out_size, void* d_ws, size_t ws_size,
                              hipStream_t stream) {
  (void)n_in; (void)out_size; (void)ws_size;
  const float* cls[3] = {(const float*)d_in[0], (const float*)d_in[2], (const float*)d_in[4]};
  const float* loc[3] = {(const float*)d_in[1], (const float*)d_in[3], (const float*)d_in[5]};
  const int hh[3]  = {80, 40, 20};
  const int wwv[3] = {80, 40, 20};
  const int str[3] = {8, 16, 32};
  int B = in_sizes[0] / (NUM_FG * A_ANCH * 80 * 80);
  if (B <= 0) B = 1;

  uint8_t* w = (uint8_t*)d_ws;
  size_t o = 0;
  auto take = [&](size_t bytes) -> void* {
    void* p = (void*)(w + o);
    o = (o + bytes + 255) & ~(size_t)255;
    return p;
  };
  uint32_t* hist1    = (uint32_t*)take((size_t)3 * B * 4096 * 4);
  uint32_t* hist2    = (uint32_t*)take((size_t)3 * B * 4096 * 4);
  uint32_t* thrInfo  = (uint32_t*)take((size_t)3 * B * 8 * 4);
  uint32_t* counters = (uint32_t*)take((size_t)3 * B * 2 * 4);
  float*    rois     = (float*)take((size_t)B * RPI * 6 * 4);
  float*    srois    = (float*)take((size_t)B * RPI * 6 * 4);

  (void)hipMemsetAsync(d_ws, 0, o, stream);

  for (int l = 0; l < 3; ++l) {
    const int N = NUM_FG * A_ANCH * hh[l] * wwv[l];
    hist_pass1<<<dim3(256, B), 256, 0, stream>>>(cls[l], hist1 + (size_t)l * B * 4096, N);
  }
  for (int l = 0; l < 3; ++l)
    radix_select<<<B, 32, 0, stream>>>(hist1 + (size_t)l * B * 4096,
                                       thrInfo + (size_t)l * B * 8, 0);
  for (int l = 0; l < 3; ++l) {
    const int N = NUM_FG * A_ANCH * hh[l] * wwv[l];
    hist_pass2<<<dim3(256, B), 256, 0, stream>>>(cls[l], thrInfo + (size_t)l * B * 8,
                                                 hist2 + (size_t)l * B * 4096, N);
  }
  for (int l = 0; l < 3; ++l)
    radix_select<<<B, 32, 0, stream>>>(hist2 + (size_t)l * B * 4096,
                                       thrInfo + (size_t)l * B * 8, 1);
  for (int l = 0; l < 3; ++l) {
    compact_decode<<<dim3(512, B), 256, 0, stream>>>(
        cls[l], loc[l], thrInfo + (size_t)l * B * 8, counters + (size_t)l * B * 2,
        rois, hh[l], wwv[l], str[l], l * PRE_NMS);
  }
  sort_rois<<<B, 1024, 0, stream>>>(rois, srois);
  nms_out<<<B, 1024, 0, stream>>>(srois, (float*)d_out);
}